// LocalConsistentLoss_25417616457985
// MI455X (gfx1250) — compile-verified
//
#include <hip/hip_runtime.h>
#include <math.h>

// ---------------------------------------------------------------------------
// mean(|x - box3x3_mean(x)|) over fp32 (8,19,512,1024).
// Memory-bound: 318.8MB @ 23.3TB/s ~= 13.7us floor. Strategy:
//   - DOUBLE-BUFFERED Tensor Data Mover staging: TDM DMA for tile i+1
//     overlaps compute of tile i (s_wait_tensorcnt <= 1 pipelining)
//   - border tiles (zero halo) via checked scalar loads, same packed layout
//   - rolling horizontal sums (4 LDS reads / pixel)
//   - deterministic two-pass reduction; wave reduction via
//     V_WMMA_F32_16X16X4_F32 (A = lane accs, B = ones)
//   - global_prefetch_b8 hint for the tile after next
// ---------------------------------------------------------------------------

typedef float v2f __attribute__((ext_vector_type(2)));
typedef float v8f __attribute__((ext_vector_type(8)));
typedef unsigned int u32x4 __attribute__((ext_vector_type(4)));
typedef int i32x4 __attribute__((ext_vector_type(4)));
typedef int i32x8 __attribute__((ext_vector_type(8)));

#define IMG_H 512
#define IMG_W 1024
#define PLANES 152                       // 8 * 19

#define TILE_W 128
#define TILE_H 32
#define LDS_W  (TILE_W + 2)              // 130
#define LDS_H  (TILE_H + 2)              // 34
#define LDS_STRIDE LDS_W                 // packed: matches TDM row packing
#define TILE_ELEMS (LDS_H * LDS_W)       // 4420 floats
#define TILE_BYTES (TILE_ELEMS * 4)      // 17680 bytes

#define TILES_X (IMG_W / TILE_W)         // 8
#define TILES_Y (IMG_H / TILE_H)         // 16
#define TILES_PER_PLANE (TILES_X * TILES_Y)        // 128
#define TOTAL_TILES (PLANES * TILES_PER_PLANE)     // 19456

#define NBLOCKS 1024
#define NTHREADS 256
#define NWAVES (NTHREADS / 32)
#define NITER (TOTAL_TILES / NBLOCKS)    // 19, exact
static_assert(TOTAL_TILES % NBLOCKS == 0, "uniform tile count per block");

#define NTOT_F 79691776.0f               // 8*19*512*1024

#define HAS_TDM __has_builtin(__builtin_amdgcn_tensor_load_to_lds)

// Wave32 sum reduction through the matrix pipe.
// A (16x4 f32): vgpr0 = v, vgpr1 = 0  =>  A[m][0]=v(lane m), A[m][2]=v(lane m+16)
// B (4x16)    : all ones
// D[m][n] = v(m) + v(m+16)  (broadcast over n)
// c[0..7] on lane l cover rows (l<16 ? 0..7 : 8..15) => local = half-wave sum,
// full wave total = local + shfl_xor(local, 16).
__device__ __forceinline__ float wave_reduce_wmma(float v) {
  v2f a; a[0] = v;    a[1] = 0.0f;
  v2f b; b[0] = 1.0f; b[1] = 1.0f;
  v8f c = {};
  c = __builtin_amdgcn_wmma_f32_16x16x4_f32(
      /*neg_a=*/false, a, /*neg_b=*/false, b,
      /*c_mod=*/(short)0, c, /*reuse_a=*/false, /*reuse_b=*/false);
  float local = (c[0] + c[1]) + (c[2] + c[3]);
  local += (c[4] + c[5]) + (c[6] + c[7]);
  return local + __shfl_xor(local, 16, 32);
}

// Begin staging `tile` into LDS buffer `buf` (0/1). Returns true if a TDM op
// was issued (one outstanding tensor op from wave 0), false if staged with
// scalar loads (published by the barrier's dscnt wait).
__device__ __forceinline__ bool start_tile_load(const float* __restrict__ x,
                                                float* __restrict__ lds,
                                                int tile, int buf, int tid) {
  const int plane = tile / TILES_PER_PLANE;
  const int t     = tile - plane * TILES_PER_PLANE;
  const int ty    = t / TILES_X;
  const int tx    = t - ty * TILES_X;
  const int row0  = ty * TILE_H;
  const int col0  = tx * TILE_W;
  const long base = (long)plane * ((long)IMG_H * IMG_W);

  const bool interior = ((unsigned)(ty - 1) < (unsigned)(TILES_Y - 2)) &&
                        ((unsigned)(tx - 1) < (unsigned)(TILES_X - 2));

#if HAS_TDM
  if (interior) {
    // Whole 34x130 halo'd tile lies inside the plane: one TDM DMA.
    // Issue from wave 0 only (tensor ops ignore EXEC and are per-wave).
    if (tid < 32) {
      const unsigned long long gaddr =
          (unsigned long long)(const void*)(x + base +
                                            (long)(row0 - 1) * IMG_W +
                                            (col0 - 1));
      u32x4 g0;
      g0[0] = 1u;                                   // count=1, no gather
      g0[1] = (unsigned)(buf * TILE_BYTES);         // lds_addr (bytes)
      g0[2] = (unsigned)(gaddr & 0xFFFFFFFFu);      // global_addr[31:0]
      g0[3] = (unsigned)((gaddr >> 32) & 0x01FFFFFFu) | (2u << 30); // type=2

      i32x8 g1;
      g1[0] = (int)(2u << 16);                      // data_size = 4 bytes
      g1[1] = (int)((unsigned)LDS_W << 16);         // tensor_dim0 = 130 (lo16)
      g1[2] = (int)((unsigned)LDS_H << 16);         // tensor_dim1 = 34  (lo16)
      g1[3] = (int)((unsigned)LDS_W << 16);         // tile_dim0 = 130
      g1[4] = LDS_H;                                // tile_dim1 = 34, tile_dim2 = 0
      g1[5] = IMG_W;                                // tensor_dim0_stride = 1024
      g1[6] = 0;
      g1[7] = 0;

      i32x4 g2 = {0, 0, 0, 0};                      // 2-D tensor: unused
      i32x4 g3 = {0, 0, 0, 0};

#if __clang_major__ >= 23
      __builtin_amdgcn_tensor_load_to_lds(g0, g1, g2, g3, g1, 0);
#else
      __builtin_amdgcn_tensor_load_to_lds(g0, g1, g2, g3, 0);
#endif
    }
    return true;
  }
#endif
  // Border tile: checked scalar loads with zero halo, packed layout identical
  // to the TDM path (row stride LDS_W, contiguous).
  float* dst = lds + buf * TILE_ELEMS;
  for (int i = tid; i < TILE_ELEMS; i += NTHREADS) {
    const int r  = i / LDS_W;          // 0..33
    const int c  = i - r * LDS_W;      // 0..129
    const int gr = row0 + r - 1;
    const int gc = col0 + c - 1;
    float v = 0.0f;
    if ((unsigned)gr < (unsigned)IMG_H && (unsigned)gc < (unsigned)IMG_W)
      v = x[base + (long)gr * IMG_W + gc];
    dst[i] = v;
  }
  return false;
}

__global__ void __launch_bounds__(NTHREADS)
box3x3_l1_partial(const float* __restrict__ x, float* __restrict__ partial) {
  // Single shared object => group-segment offset 0, which is what the TDM
  // descriptor's lds_addr refers to. Two tile buffers + reduction scratch.
  __shared__ float smem[2 * TILE_ELEMS + NWAVES];
  float* red = smem + 2 * TILE_ELEMS;

  const int tid = threadIdx.x;
  float acc = 0.0f;

  // Prologue: stage first tile into buffer 0.
  bool cur_tdm = start_tile_load(x, smem, blockIdx.x, 0, tid);
  int p = 0;

  for (int i = 0; i < NITER; ++i) {
    const int tile = blockIdx.x + i * NBLOCKS;

    // Prefetch hint (L2) for the tile after next.
    {
      const int nt = tile + 2 * NBLOCKS;
      if (nt < TOTAL_TILES) {
        const int np  = nt / TILES_PER_PLANE;
        const int t2  = nt - np * TILES_PER_PLANE;
        const int nty = t2 / TILES_X;
        const int ntx = t2 - nty * TILES_X;
        const float* pf = x + (long)np * ((long)IMG_H * IMG_W)
                            + (long)(nty * TILE_H) * IMG_W + ntx * TILE_W + tid;
        __builtin_prefetch(pf, 0, 1);   // -> global_prefetch_b8
      }
    }

    // Kick off the next tile's DMA into the other buffer (overlaps compute).
    bool next_tdm = false;
    if (i + 1 < NITER)
      next_tdm = start_tile_load(x, smem, tile + NBLOCKS, p ^ 1, tid);

    // Current buffer ready: TDM ops from wave 0 complete in order, so
    // "<= 1 outstanding" means tile i has landed while tile i+1 streams.
    if (cur_tdm && tid < 32) {
      if (next_tdm) __builtin_amdgcn_s_wait_tensorcnt(1);
      else          __builtin_amdgcn_s_wait_tensorcnt(0);
    }
    __syncthreads();

    // Each thread: one column, 16 consecutive rows; rolling horizontal sums.
    const float* lds = smem + p * TILE_ELEMS;
    const int cx = tid & (TILE_W - 1);           // 0..127
    const int ry = (tid >> 7) * (TILE_H / 2);    // 0 or 16
    int o = ry * LDS_STRIDE + cx;
    float h0 = lds[o] + lds[o + 1] + lds[o + 2];
    o += LDS_STRIDE;
    float h1 = lds[o] + lds[o + 1] + lds[o + 2];
#pragma unroll
    for (int q = 0; q < TILE_H / 2; ++q) {
      o += LDS_STRIDE;
      const float h2 = lds[o] + lds[o + 1] + lds[o + 2];
      const float center = lds[o - LDS_STRIDE + 1];
      acc += fabsf(fmaf(-(h0 + h1 + h2), (1.0f / 9.0f), center));
      h0 = h1; h1 = h2;
    }
    // Buffer p fully consumed before iteration i+1 re-targets it.
    __syncthreads();

    cur_tdm = next_tdm;
    p ^= 1;
  }

  // Deterministic block reduction: WMMA wave reduce -> 8 partials -> lane 0.
  const float wtot = wave_reduce_wmma(acc);
  const int lane = tid & 31, wid = tid >> 5;
  if (lane == 0) red[wid] = wtot;
  __syncthreads();
  if (tid == 0) {
    float s = 0.0f;
#pragma unroll
    for (int i = 0; i < NWAVES; ++i) s += red[i];
    partial[blockIdx.x] = s;
  }
}

__global__ void __launch_bounds__(NTHREADS)
box3x3_l1_final(const float* __restrict__ partial, float* __restrict__ out, int n) {
  __shared__ float red[NWAVES];
  const int tid = threadIdx.x;
  float acc = 0.0f;
  for (int i = tid; i < n; i += NTHREADS) acc += partial[i];

  const float wtot = wave_reduce_wmma(acc);
  const int lane = tid & 31, wid = tid >> 5;
  if (lane == 0) red[wid] = wtot;
  __syncthreads();
  if (tid == 0) {
    float s = 0.0f;
#pragma unroll
    for (int i = 0; i < NWAVES; ++i) s += red[i];
    out[0] = s * (1.0f / NTOT_F);
  }
}

extern "C" void kernel_launch(void* const* d_in, const int* in_sizes, int n_in,
                              void* d_out, int out_size, void* d_ws, size_t ws_size,
                              hipStream_t stream) {
  (void)in_sizes; (void)n_in; (void)out_size; (void)ws_size;
  const float* x  = (const float*)d_in[0];
  float* out      = (float*)d_out;
  float* partials = (float*)d_ws;          // needs NBLOCKS * 4 = 4 KB

  box3x3_l1_partial<<<NBLOCKS, NTHREADS, 0, stream>>>(x, partials);
  box3x3_l1_final<<<1, NTHREADS, 0, stream>>>(partials, out, NBLOCKS);
}